// simple_gnn_gcn_17781164605885
// MI455X (gfx1250) — compile-verified
//
#include <hip/hip_runtime.h>

typedef float v2f __attribute__((ext_vector_type(2)));
typedef float v8f __attribute__((ext_vector_type(8)));

#define FIN 20   // input feature dim
#define F1  16   // hidden dim

// ---------------------------------------------------------------------------
// Kernel 1: per-wave 16-node tile.  WMMA f32 16x16x4, K=20 in 5 chunks.
//   y1  = x @ W_rel1.T            [N,16]   (to be edge-aggregated)
//   r1  = x @ W_root1.T + b_rel1  [N,16]   (per-node residual term)
//   agg1 = 0                               (atomic accumulator init)
// ---------------------------------------------------------------------------
__global__ void k_transform1(const float* __restrict__ x,
                             const float* __restrict__ W_rel1,
                             const float* __restrict__ b_rel1,
                             const float* __restrict__ W_root1,
                             float* __restrict__ y1,
                             float* __restrict__ r1,
                             float* __restrict__ agg1,
                             int n_nodes, int n_tiles)
{
    const int lane = threadIdx.x & 31;
    const int wave = threadIdx.x >> 5;
    const int tile = blockIdx.x * (blockDim.x >> 5) + wave;
    if (tile >= n_tiles) return;          // whole-wave uniform branch

    const int half = lane >> 4;           // 0: lanes 0-15, 1: lanes 16-31
    const int n    = lane & 15;           // output column (and A-row id)
    const int base = tile * 16;

    v8f c_rel  = {};
    v8f c_root = {};

    #pragma unroll
    for (int kk = 0; kk < 5; ++kk) {
        const int k0 = kk * 4 + half * 2;
        // A fragment: x[base+n][k0], x[base+n][k0+1]   (clamped, keeps EXEC full)
        int mrow = base + n;
        if (mrow >= n_nodes) mrow = n_nodes - 1;
        v2f a = *(const v2f*)(x + (size_t)mrow * FIN + k0);
        // B fragments: B[k][col] = W[col][k]
        v2f brel  = *(const v2f*)(W_rel1  + n * FIN + k0);
        v2f broot = *(const v2f*)(W_root1 + n * FIN + k0);
        c_rel  = __builtin_amdgcn_wmma_f32_16x16x4_f32(false, a, false, brel,
                                                       (short)0, c_rel,  false, false);
        c_root = __builtin_amdgcn_wmma_f32_16x16x4_f32(false, a, false, broot,
                                                       (short)0, c_root, false, false);
    }

    const float bias = b_rel1[n];
    if (base + 16 <= n_nodes) {
        // Full tile (always the case for N % 16 == 0): branchless epilogue.
        #pragma unroll
        for (int j = 0; j < 8; ++j) {
            const size_t idx = (size_t)(base + j + 8 * half) * F1 + n;
            y1  [idx] = c_rel[j];
            r1  [idx] = c_root[j] + bias;
            agg1[idx] = 0.0f;
        }
    } else {
        #pragma unroll
        for (int j = 0; j < 8; ++j) {
            const int row = base + j + 8 * half;
            if (row < n_nodes) {
                const size_t idx = (size_t)row * F1 + n;
                y1  [idx] = c_rel[j];
                r1  [idx] = c_root[j] + bias;
                agg1[idx] = 0.0f;
            }
        }
    }
}

// ---------------------------------------------------------------------------
// Kernel 2: edge scatter, layer 1.  16 consecutive lanes = one edge, one
// feature each -> coalesced 64B gather of y1[src] and 64B of atomics.
// ---------------------------------------------------------------------------
__global__ void k_scatter1(const int* __restrict__ src,
                           const int* __restrict__ dst,
                           const float* __restrict__ y1,
                           float* __restrict__ agg1,
                           long long n_edges)
{
    const long long gid = (long long)blockIdx.x * blockDim.x + threadIdx.x;
    const long long e   = gid >> 4;
    const int       f   = (int)(gid & 15);
    if (e >= n_edges) return;
    const int s = src[e];
    const int d = dst[e];
    unsafeAtomicAdd(&agg1[(size_t)d * F1 + f], y1[(size_t)s * F1 + f]);
}

// ---------------------------------------------------------------------------
// Kernel 3: h = relu(agg1 + r1);  s2 = h . w_rel2;  out = h . w_root2 + b_rel2
// (out doubles as the layer-2 accumulator, pre-loaded with the root term)
// ---------------------------------------------------------------------------
__global__ void k_relu_s2(const float* __restrict__ agg1,
                          const float* __restrict__ r1,
                          const float* __restrict__ W_rel2,
                          const float* __restrict__ b_rel2,
                          const float* __restrict__ W_root2,
                          float* __restrict__ s2,
                          float* __restrict__ out,
                          int n_nodes)
{
    const int nidx = blockIdx.x * blockDim.x + threadIdx.x;
    if (nidx >= n_nodes) return;

    const float4* a4 = (const float4*)(agg1 + (size_t)nidx * F1);
    const float4* r4 = (const float4*)(r1   + (size_t)nidx * F1);

    float srel = 0.0f, sroot = 0.0f;
    #pragma unroll
    for (int q = 0; q < 4; ++q) {
        float4 av = a4[q];
        float4 rv = r4[q];
        float h0 = fmaxf(av.x + rv.x, 0.0f);
        float h1 = fmaxf(av.y + rv.y, 0.0f);
        float h2 = fmaxf(av.z + rv.z, 0.0f);
        float h3 = fmaxf(av.w + rv.w, 0.0f);
        const int i = q * 4;
        srel  += h0 * W_rel2[i]     + h1 * W_rel2[i + 1]
               + h2 * W_rel2[i + 2] + h3 * W_rel2[i + 3];
        sroot += h0 * W_root2[i]     + h1 * W_root2[i + 1]
               + h2 * W_root2[i + 2] + h3 * W_root2[i + 3];
    }
    s2 [nidx] = srel;
    out[nidx] = sroot + b_rel2[0];
}

// ---------------------------------------------------------------------------
// Kernel 4: edge scatter, layer 2.  One scalar atomic per edge.
// ---------------------------------------------------------------------------
__global__ void k_scatter2(const int* __restrict__ src,
                           const int* __restrict__ dst,
                           const float* __restrict__ s2,
                           float* __restrict__ out,
                           long long n_edges)
{
    const long long e = (long long)blockIdx.x * blockDim.x + threadIdx.x;
    if (e >= n_edges) return;
    unsafeAtomicAdd(&out[dst[e]], s2[src[e]]);
}

// ---------------------------------------------------------------------------
extern "C" void kernel_launch(void* const* d_in, const int* in_sizes, int n_in,
                              void* d_out, int out_size, void* d_ws, size_t ws_size,
                              hipStream_t stream)
{
    (void)n_in; (void)out_size; (void)ws_size;

    const float* x       = (const float*)d_in[0];
    const int*   ei      = (const int*)  d_in[1];   // [2, E] int32
    const float* W_rel1  = (const float*)d_in[2];   // [16,20]
    const float* b_rel1  = (const float*)d_in[3];   // [16]
    const float* W_root1 = (const float*)d_in[4];   // [16,20]
    const float* W_rel2  = (const float*)d_in[5];   // [1,16]
    const float* b_rel2  = (const float*)d_in[6];   // [1]
    const float* W_root2 = (const float*)d_in[7];   // [1,16]

    const int       N = in_sizes[0] / FIN;          // 100000
    const long long E = in_sizes[1] / 2;            // 3200000
    const int* src = ei;
    const int* dst = ei + E;

    // Workspace layout (floats): y1[N*16] | r1[N*16] | agg1[N*16] | s2[N]
    float* y1   = (float*)d_ws;
    float* r1   = y1   + (size_t)N * F1;
    float* agg1 = r1   + (size_t)N * F1;
    float* s2   = agg1 + (size_t)N * F1;
    float* out  = (float*)d_out;

    // 1) WMMA node transforms + agg1 zero-init (2 waves / 2 tiles per block)
    const int tiles = (N + 15) / 16;
    k_transform1<<<dim3((tiles + 1) / 2), dim3(64), 0, stream>>>(
        x, W_rel1, b_rel1, W_root1, y1, r1, agg1, N, tiles);

    // 2) Layer-1 edge aggregation: thread per (edge, feature)
    const long long t1 = E * F1;
    k_scatter1<<<dim3((unsigned)((t1 + 255) / 256)), dim3(256), 0, stream>>>(
        src, dst, y1, agg1, E);

    // 3) ReLU + layer-2 node transforms (scalar per node), out pre-init
    k_relu_s2<<<dim3((N + 255) / 256), dim3(256), 0, stream>>>(
        agg1, r1, W_rel2, b_rel2, W_root2, s2, out, N);

    // 4) Layer-2 edge aggregation: scalar atomic per edge
    k_scatter2<<<dim3((unsigned)((E + 255) / 256)), dim3(256), 0, stream>>>(
        src, dst, s2, out, E);
}